// GCN_Dense_Att_7627861917713
// MI455X (gfx1250) — compile-verified
//
#include <hip/hip_runtime.h>
#include <math.h>
#include <stdint.h>

// ---------------------------------------------------------------------------
// Problem constants (match reference)
// ---------------------------------------------------------------------------
#define NODES 50000
#define DREL  3
#define EDGES 600000
#define IN_C  300
#define HID_C 128   // == OUT_C == 128 channels everywhere after layer 1

typedef __attribute__((ext_vector_type(2))) float v2f;
typedef __attribute__((ext_vector_type(8))) float v8f;

// ---------------------------------------------------------------------------
// CDNA5 async global->LDS path (ASYNCcnt-tracked DMA, no VGPR round trip).
// Guarded so the file compiles on toolchains without the builtins (and in the
// host pass, where __has_builtin is false for device builtins).
//
// Probe-derived signature (clang diagnostic round 2):
//   __builtin_amdgcn_global_load_async_to_lds_b128(
//       int4-addrspace(1)* gsrc, int4-addrspace(3)* ldst, imm offset, imm cpol)
// ---------------------------------------------------------------------------
#if defined(__has_builtin)
#  if __has_builtin(__builtin_amdgcn_global_load_async_to_lds_b128) && \
      __has_builtin(__builtin_amdgcn_s_wait_asynccnt)
#    define CDNA5_ASYNC_LDS 1
#  endif
#endif
#ifndef CDNA5_ASYNC_LDS
#  define CDNA5_ASYNC_LDS 0
#endif

#if CDNA5_ASYNC_LDS
typedef int v4i_vs __attribute__((vector_size(16)));
typedef __attribute__((address_space(1))) v4i_vs* g_b128_ptr;   // global int4*
typedef __attribute__((address_space(3))) v4i_vs* l_b128_ptr;   // LDS int4*
#endif

__device__ __forceinline__ void copy16_to_lds(const float* __restrict__ src,
                                              float* __restrict__ lds_dst)
{
#if CDNA5_ASYNC_LDS
    // global_load_async_to_lds_b128: 16B global -> 16B LDS, tracked on ASYNCcnt.
    // LDS operand uses the low 32 bits of the generic address (ISA aperture rule).
    __builtin_amdgcn_global_load_async_to_lds_b128(
        (g_b128_ptr)(uintptr_t)src,
        (l_b128_ptr)(uint32_t)(uintptr_t)lds_dst,
        /*imm offset=*/0, /*cpol=*/0);
#else
    *(float4*)lds_dst = *(const float4*)src;
#endif
}

__device__ __forceinline__ void wait_lds_fill()
{
#if CDNA5_ASYNC_LDS
    __builtin_amdgcn_s_wait_asynccnt(0);   // s_wait_asynccnt 0
#endif
    __syncthreads();
}

// ---------------------------------------------------------------------------
// Dense GEMM with bias: C[M,128] = A[M,K] @ W[K,128] + bias[128]
// One wave -> one 16-row M strip x all 128 columns (8 WMMA accumulators),
// V_WMMA_F32_16X16X4_F32 (exact f32 semantics, matches reference).
//
// W is staged per-block into LDS in k-chunks of KC=100 rows via the async
// global->LDS engine; padded LDS row stride 136 floats makes the half-wave
// B-fragment reads bank-conflict-free (2*136 mod 64 == 16 -> disjoint
// 16-bank windows). Inner loop: 1 global b64 (A) + 16 ds b32 (B) feeding
// 8 WMMAs per k-step.
//
// A-frag (16x4 f32): lanes 0-15 hold M=0..15; VGPR0 = K {0|2}, VGPR1 = K {1|3}
//                    selected by lane>>4. B-frag (4x16) mirrors this layout.
// C-frag (16x16 f32, 8 VGPRs): VGPR v -> M = v + 8*(lane>>4), N = lane&15.
// ---------------------------------------------------------------------------
template <int K>
__global__ __launch_bounds__(256) void wmma_gemm_bias(
    const float* __restrict__ A, const float* __restrict__ W,
    const float* __restrict__ bias, float* __restrict__ C, int M)
{
    constexpr int KC  = 100;          // k-rows per LDS chunk (300 = 3*100, 128 = 100+28)
    constexpr int LDW = 136;          // padded LDS row stride (floats)
    __shared__ float ldsW[KC * LDW];  // 54.4 KB

    const int wave   = threadIdx.x >> 5;
    const int lane   = threadIdx.x & 31;
    const int mtiles = M / 16;
    const int mtile  = blockIdx.x * 8 + wave;            // 16-row strip index
    const int mt     = (mtile < mtiles) ? mtile : (mtiles - 1);  // clamp: keep all
    const bool live  = (mtile < mtiles);                 // waves stay for barriers

    const int lane15 = lane & 15;
    const int kOff   = (lane >> 4) * 2;                  // {0,2}
    const float* __restrict__ arow = A + (size_t)(mt * 16 + lane15) * K;

    v8f acc[8];
#pragma unroll
    for (int nt = 0; nt < 8; ++nt) acc[nt] = (v8f){0.f,0.f,0.f,0.f,0.f,0.f,0.f,0.f};

    for (int k0 = 0; k0 < K; k0 += KC) {
        const int kc = (K - k0 < KC) ? (K - k0) : KC;    // 100 or 28 (both %4==0)

        __syncthreads();                                 // protect LDS reuse
        // Cooperative fill: kc*32 float4 chunks, fully coalesced both sides.
        for (int c = threadIdx.x; c < kc * 32; c += 256) {
            const int krow = c >> 5;                     // row within chunk
            const int cg   = (c & 31) * 4;               // column group
            copy16_to_lds(W + (size_t)(k0 + krow) * 128 + cg,
                          &ldsW[krow * LDW + cg]);
        }
        wait_lds_fill();

        for (int k = 0; k < kc; k += 4) {
            v2f a;
            a.x = arow[k0 + k + kOff];
            a.y = arow[k0 + k + kOff + 1];
#pragma unroll
            for (int nt = 0; nt < 8; ++nt) {
                const int col = nt * 16 + lane15;
                v2f b;
                b.x = ldsW[(k + kOff)     * LDW + col];
                b.y = ldsW[(k + kOff + 1) * LDW + col];
                acc[nt] = __builtin_amdgcn_wmma_f32_16x16x4_f32(
                    /*neg_a=*/false, a, /*neg_b=*/false, b,
                    /*c_mod=*/(short)0, acc[nt], /*reuse_a=*/false, /*reuse_b=*/false);
            }
        }
    }

    if (live) {
        const int rowBase = mt * 16 + (lane >> 4) * 8;
#pragma unroll
        for (int nt = 0; nt < 8; ++nt) {
            const float bv  = bias[nt * 16 + lane15];
            const int   col = nt * 16 + lane15;
#pragma unroll
            for (int v = 0; v < 8; ++v)
                C[(size_t)(rowBase + v) * 128 + col] = acc[nt][v] + bv;
        }
    }
}

// ---------------------------------------------------------------------------
// Zero a float buffer (graph-capture safe; no hipMemset)
// ---------------------------------------------------------------------------
__global__ void zero_f32(float* __restrict__ p, long n)
{
    long i = (long)blockIdx.x * blockDim.x + threadIdx.x;
    long stride = (long)gridDim.x * blockDim.x;
    for (; i < n; i += stride) p[i] = 0.0f;
}

// ---------------------------------------------------------------------------
// Edge aggregation: out[rows[e]] += softmax(att)[rel] * edge_w[e] * support[cols[e]]
// One wave per edge: 128 channels = 32 lanes x float4 (global_load_b128 gather,
// global_atomic_add_f32 scatter). 25.6 MB tables live in the 192 MB L2.
// ---------------------------------------------------------------------------
__global__ __launch_bounds__(256) void aggregate_edges(
    const float* __restrict__ support,   // [NODES,128]
    const int*   __restrict__ rows,      // [DREL*EDGES]
    const int*   __restrict__ cols,      // [DREL*EDGES]
    const float* __restrict__ edge_w,    // [DREL*EDGES]
    const float* __restrict__ att,       // [DREL]
    float*       __restrict__ out)       // [NODES,128] (pre-zeroed)
{
    const long gw   = (long)blockIdx.x * 8 + (threadIdx.x >> 5);  // global wave id
    const int  lane = threadIdx.x & 31;
    const long total = (long)DREL * EDGES;
    if (gw >= total) return;

    const int rel = (int)(gw / EDGES);

    // softmax over 3 attention logits (tiny, recomputed per wave)
    const float a0 = att[0], a1 = att[1], a2 = att[2];
    const float mx = fmaxf(a0, fmaxf(a1, a2));
    const float e0 = expf(a0 - mx), e1 = expf(a1 - mx), e2 = expf(a2 - mx);
    const float inv = 1.0f / (e0 + e1 + e2);
    const float arel = (rel == 0 ? e0 : (rel == 1 ? e1 : e2)) * inv;

    const float coef = arel * edge_w[gw];
    const int   col  = cols[gw];
    const int   row  = rows[gw];

    const float4 v = *(const float4*)(support + (size_t)col * 128 + lane * 4);
    float* dst = out + (size_t)row * 128 + lane * 4;
    atomicAdd(dst + 0, coef * v.x);
    atomicAdd(dst + 1, coef * v.y);
    atomicAdd(dst + 2, coef * v.z);
    atomicAdd(dst + 3, coef * v.w);
}

// ---------------------------------------------------------------------------
// LeakyReLU(0.2) in place, vectorized float4
// ---------------------------------------------------------------------------
__global__ void leaky_relu(float* __restrict__ p, long n4)
{
    long i = (long)blockIdx.x * blockDim.x + threadIdx.x;
    long stride = (long)gridDim.x * blockDim.x;
    for (; i < n4; i += stride) {
        float4 v = ((float4*)p)[i];
        v.x = v.x >= 0.f ? v.x : 0.2f * v.x;
        v.y = v.y >= 0.f ? v.y : 0.2f * v.y;
        v.z = v.z >= 0.f ? v.z : 0.2f * v.z;
        v.w = v.w >= 0.f ? v.w : 0.2f * v.w;
        ((float4*)p)[i] = v;
    }
}

// ---------------------------------------------------------------------------
// L2 row normalization in place: one wave per 128-channel row.
// ---------------------------------------------------------------------------
__global__ __launch_bounds__(256) void l2_normalize(float* __restrict__ h, int n)
{
    const int row  = blockIdx.x * 8 + (threadIdx.x >> 5);
    const int lane = threadIdx.x & 31;
    if (row >= n) return;

    float4 v = *(float4*)(h + (size_t)row * 128 + lane * 4);
    float s = v.x * v.x + v.y * v.y + v.z * v.z + v.w * v.w;
#pragma unroll
    for (int m = 16; m >= 1; m >>= 1) s += __shfl_xor(s, m, 32);

    const float inv = 1.0f / fmaxf(sqrtf(s), 1e-12f);
    v.x *= inv; v.y *= inv; v.z *= inv; v.w *= inv;
    *(float4*)(h + (size_t)row * 128 + lane * 4) = v;
}

// ---------------------------------------------------------------------------
// Orchestration
// ---------------------------------------------------------------------------
extern "C" void kernel_launch(void* const* d_in, const int* in_sizes, int n_in,
                              void* d_out, int out_size, void* d_ws, size_t ws_size,
                              hipStream_t stream)
{
    const float* x      = (const float*)d_in[0];   // [N, 300]
    const float* edge_w = (const float*)d_in[1];   // [D, E]
    const float* w1     = (const float*)d_in[2];   // [300, 128]
    const float* b1     = (const float*)d_in[3];   // [128]
    const float* a1     = (const float*)d_in[4];   // [3]
    const float* w2     = (const float*)d_in[5];   // [128, 128]
    const float* b2     = (const float*)d_in[6];   // [128]
    const float* a2     = (const float*)d_in[7];   // [3]
    const int*   rows   = (const int*)d_in[8];     // [D, E]
    const int*   cols   = (const int*)d_in[9];     // [D, E]
    float*       out    = (float*)d_out;           // [N, 128]

    float* support = (float*)d_ws;                       // N*128 floats
    float* hbuf    = support + (size_t)NODES * 128;      // N*128 floats

    const long feat_n  = (long)NODES * 128;
    const int  mtiles  = NODES / 16;                     // 3125 (exact)
    const int  gemm_gr = (mtiles + 7) / 8;               // 8 waves/block
    const long nwaves  = (long)DREL * EDGES;             // 1.8M edge-waves
    const int  agg_gr  = (int)((nwaves + 7) / 8);

    // ---- Layer 1 ----
    wmma_gemm_bias<IN_C><<<gemm_gr, 256, 0, stream>>>(x, w1, b1, support, NODES);
    zero_f32<<<2048, 256, 0, stream>>>(hbuf, feat_n);
    aggregate_edges<<<agg_gr, 256, 0, stream>>>(support, rows, cols, edge_w, a1, hbuf);
    leaky_relu<<<2048, 256, 0, stream>>>(hbuf, feat_n / 4);

    // ---- Layer 2 ----
    wmma_gemm_bias<HID_C><<<gemm_gr, 256, 0, stream>>>(hbuf, w2, b2, support, NODES);
    zero_f32<<<2048, 256, 0, stream>>>(out, feat_n);
    aggregate_edges<<<agg_gr, 256, 0, stream>>>(support, rows, cols, edge_w, a2, out);

    // ---- F.normalize ----
    l2_normalize<<<(NODES + 7) / 8, 256, 0, stream>>>(out, NODES);
}